// GAT_62259845923392
// MI455X (gfx1250) — compile-verified
//
#include <hip/hip_runtime.h>
#include <hip/hip_bf16.h>

// ---------------------------------------------------------------------------
// GAT (2x GATConv, heads=1) for MI455X / gfx1250.
// GEMMs use V_WMMA_F32_16X16X4_F32 (full fp32 precision, wave32 WMMA).
// Aggregation is L2-resident (total working set ~70MB << 192MB L2).
// ---------------------------------------------------------------------------

typedef float v2f __attribute__((ext_vector_type(2)));
typedef float v8f __attribute__((ext_vector_type(8)));

#define NEG_SLOPE 0.2f

__device__ __forceinline__ float lrelu(float x) {
    return x > 0.0f ? x : NEG_SLOPE * x;
}

// Monotone float -> uint encoding so atomicMax(u32) implements float max.
__device__ __forceinline__ unsigned enc_f32(float f) {
    unsigned u = __float_as_uint(f);
    return (u & 0x80000000u) ? ~u : (u | 0x80000000u);
}
__device__ __forceinline__ float dec_f32(unsigned e) {
    return (e & 0x80000000u) ? __uint_as_float(e & 0x7FFFFFFFu)
                             : __uint_as_float(~e);
}

// edge e: real edges [0,Eg), then self loops [Eg, Eg+N)
__device__ __forceinline__ void edge_sd(const int* __restrict__ ei, int Eg,
                                        int e, int& s, int& d) {
    if (e < Eg) { s = ei[e]; d = ei[Eg + e]; }
    else        { s = e - Eg; d = e - Eg; }
}

// ---------------------------------------------------------------------------
// WMMA GEMM: H[nrows, NOUT] = A[nrows, 128] @ W[128, NOUT]
// One wave owns a 16-row strip; NT = NOUT/16 accumulators held in VGPRs;
// K stepped by 4 using V_WMMA_F32_16X16X4_F32.
//
// f32 16x4 A fragment (2 VGPRs): lanes 0-15 hold M=lane, v0=K+0, v1=K+1;
//                                lanes 16-31 hold M=lane-16, v0=K+2, v1=K+3.
// f32 4x16 B fragment (2 VGPRs): mirrored (rows K+0/K+1 low half, K+2/K+3 high).
// f32 16x16 C/D: VGPR r -> row (r + 8*hi), col = lane&15.
// nrows must be a multiple of 16 (50000 = 16*3125: exact).
// ---------------------------------------------------------------------------
template <int NOUT>
__global__ void __launch_bounds__(256)
gat_gemm_wmma(const float* __restrict__ A, const float* __restrict__ W,
              float* __restrict__ H, int nrows) {
    constexpr int KDIM = 128;
    constexpr int NT = NOUT / 16;
    const int wave = threadIdx.x >> 5;
    const int lane = threadIdx.x & 31;
    const int tile = blockIdx.x * 8 + wave;
    const int m0 = tile * 16;
    if (m0 >= nrows) return;                 // wave-uniform exit
    const int lm = lane & 15;
    const bool hi = lane >= 16;

    v8f acc[NT] = {};
    const float* __restrict__ arow = A + (size_t)(m0 + lm) * KDIM;

    for (int k0 = 0; k0 < KDIM; k0 += 4) {
        const int k = k0 + (hi ? 2 : 0);
        v2f afrag;
        afrag.x = arow[k];
        afrag.y = arow[k + 1];
#pragma unroll
        for (int t = 0; t < NT; ++t) {
            v2f bfrag;
            bfrag.x = W[(size_t)k * NOUT + t * 16 + lm];
            bfrag.y = W[(size_t)(k + 1) * NOUT + t * 16 + lm];
            acc[t] = __builtin_amdgcn_wmma_f32_16x16x4_f32(
                false, afrag, false, bfrag, (short)0, acc[t], false, false);
        }
    }

#pragma unroll
    for (int t = 0; t < NT; ++t) {
#pragma unroll
        for (int r = 0; r < 8; ++r) {
            const int row = m0 + r + (hi ? 8 : 0);
            H[(size_t)row * NOUT + t * 16 + lm] = acc[t][r];
        }
    }
}

// ---------------------------------------------------------------------------
// Per-node attention logits: asrc[i] = h[i]·a_src, adst[i] = h[i]·a_dst.
// One wave per node, wave32 shuffle reduction.
// ---------------------------------------------------------------------------
__global__ void __launch_bounds__(256)
gat_alpha(const float* __restrict__ H, int nrows, int nfeat,
          const float* __restrict__ a_src, const float* __restrict__ a_dst,
          float* __restrict__ asrc, float* __restrict__ adst) {
    const int wave = threadIdx.x >> 5;
    const int lane = threadIdx.x & 31;
    const int node = blockIdx.x * (blockDim.x >> 5) + wave;
    if (node >= nrows) return;
    float s = 0.0f, d = 0.0f;
    for (int c = lane; c < nfeat; c += 32) {
        const float h = H[(size_t)node * nfeat + c];
        s += h * a_src[c];
        d += h * a_dst[c];
    }
#pragma unroll
    for (int off = 16; off > 0; off >>= 1) {
        s += __shfl_xor(s, off, 32);
        d += __shfl_xor(d, off, 32);
    }
    if (lane == 0) { asrc[node] = s; adst[node] = d; }
}

// ---------------------------------------------------------------------------
// Fills
// ---------------------------------------------------------------------------
__global__ void fill_f32(float* __restrict__ p, float v, int n) {
    const int i = blockIdx.x * blockDim.x + threadIdx.x;
    if (i < n) p[i] = v;
}
__global__ void fill_u32(unsigned* __restrict__ p, unsigned v, int n) {
    const int i = blockIdx.x * blockDim.x + threadIdx.x;
    if (i < n) p[i] = v;
}

// ---------------------------------------------------------------------------
// Segment softmax passes over edges
// ---------------------------------------------------------------------------
__global__ void __launch_bounds__(256)
edge_max(const int* __restrict__ ei, int Eg, int Etot,
         const float* __restrict__ asrc, const float* __restrict__ adst,
         unsigned* __restrict__ emax_enc) {
    const int e = blockIdx.x * blockDim.x + threadIdx.x;
    if (e >= Etot) return;
    int s, d;
    edge_sd(ei, Eg, e, s, d);
    const float v = lrelu(asrc[s] + adst[d]);
    atomicMax(&emax_enc[d], enc_f32(v));
}

__global__ void __launch_bounds__(256)
edge_denom(const int* __restrict__ ei, int Eg, int Etot,
           const float* __restrict__ asrc, const float* __restrict__ adst,
           const unsigned* __restrict__ emax_enc, float* __restrict__ denom) {
    const int e = blockIdx.x * blockDim.x + threadIdx.x;
    if (e >= Etot) return;
    int s, d;
    edge_sd(ei, Eg, e, s, d);
    const float v = lrelu(asrc[s] + adst[d]);
    atomicAdd(&denom[d], expf(v - dec_f32(emax_enc[d])));
}

// One wave per edge: OUT[d] += alpha * H[s]  (NF/32 atomics per lane)
template <int NF>
__global__ void __launch_bounds__(256)
edge_aggr(const int* __restrict__ ei, int Eg, int Etot,
          const float* __restrict__ asrc, const float* __restrict__ adst,
          const unsigned* __restrict__ emax_enc, const float* __restrict__ denom,
          const float* __restrict__ H, float* __restrict__ OUT) {
    const int wave = threadIdx.x >> 5;
    const int lane = threadIdx.x & 31;
    const int e = blockIdx.x * (blockDim.x >> 5) + wave;
    if (e >= Etot) return;
    int s, d;
    edge_sd(ei, Eg, e, s, d);
    const float v = lrelu(asrc[s] + adst[d]);
    const float alpha = expf(v - dec_f32(emax_enc[d])) / denom[d];
#pragma unroll
    for (int c = lane; c < NF; c += 32)
        atomicAdd(&OUT[(size_t)d * NF + c], alpha * H[(size_t)s * NF + c]);
}

// X = relu(X + b)  (layer-1 epilogue)
__global__ void bias_relu(float* __restrict__ X, const float* __restrict__ b,
                          int total, int nf) {
    const int i = blockIdx.x * blockDim.x + threadIdx.x;
    if (i >= total) return;
    const float v = X[i] + b[i % nf];
    X[i] = v > 0.0f ? v : 0.0f;
}

// out = log_softmax(AGG + b) over 32 classes; one wave per node, lane==class.
__global__ void __launch_bounds__(256)
logsoftmax32(const float* __restrict__ AGG, const float* __restrict__ b,
             float* __restrict__ out, int nrows) {
    const int wave = threadIdx.x >> 5;
    const int lane = threadIdx.x & 31;
    const int node = blockIdx.x * (blockDim.x >> 5) + wave;
    if (node >= nrows) return;
    const float z = AGG[(size_t)node * 32 + lane] + b[lane];
    float m = z;
#pragma unroll
    for (int off = 16; off > 0; off >>= 1) m = fmaxf(m, __shfl_xor(m, off, 32));
    const float ex = expf(z - m);
    float ssum = ex;
#pragma unroll
    for (int off = 16; off > 0; off >>= 1) ssum += __shfl_xor(ssum, off, 32);
    out[(size_t)node * 32 + lane] = z - m - logf(ssum);
}

// ---------------------------------------------------------------------------
// Host-side orchestration
// ---------------------------------------------------------------------------
static inline int cdiv(int a, int b) { return (a + b - 1) / b; }

extern "C" void kernel_launch(void* const* d_in, const int* in_sizes, int n_in,
                              void* d_out, int out_size, void* d_ws, size_t ws_size,
                              hipStream_t stream) {
    const float* x   = (const float*)d_in[0];
    const int*   ei  = (const int*)  d_in[1];   // [2, Eg] flat, int32
    const float* W1  = (const float*)d_in[2];
    const float* as1 = (const float*)d_in[3];
    const float* ad1 = (const float*)d_in[4];
    const float* b1  = (const float*)d_in[5];
    const float* W2  = (const float*)d_in[6];
    const float* as2 = (const float*)d_in[7];
    const float* ad2 = (const float*)d_in[8];
    const float* b2  = (const float*)d_in[9];
    float* out = (float*)d_out;

    const int N  = in_sizes[0] / 128;   // 50000 (multiple of 16)
    const int Eg = in_sizes[1] / 2;     // 800000
    const int Et = Eg + N;              // + self loops

    // Workspace carve-out
    float* ws = (float*)d_ws;
    float* h1   = ws; ws += (size_t)N * 128;
    float* agg1 = ws; ws += (size_t)N * 128;
    float* h2   = ws; ws += (size_t)N * 32;
    float* agg2 = ws; ws += (size_t)N * 32;
    float*    asrc1 = ws;             ws += N;
    float*    adst1 = ws;             ws += N;
    float*    den1  = ws;             ws += N;
    unsigned* emx1  = (unsigned*)ws;  ws += N;
    float*    asrc2 = ws;             ws += N;
    float*    adst2 = ws;             ws += N;
    float*    den2  = ws;             ws += N;
    unsigned* emx2  = (unsigned*)ws;  ws += N;

    const unsigned ENC_NEG_INF = ~0xFF800000u;   // enc(-inf)
    const int B = 256;
    const int wavesPerBlk = B / 32;

    // ---- Layer 1 ----
    gat_gemm_wmma<128><<<cdiv(N / 16, wavesPerBlk), B, 0, stream>>>(x, W1, h1, N);
    gat_alpha<<<cdiv(N, wavesPerBlk), B, 0, stream>>>(h1, N, 128, as1, ad1, asrc1, adst1);

    fill_f32<<<cdiv(N * 128, B), B, 0, stream>>>(agg1, 0.0f, N * 128);
    fill_f32<<<cdiv(N, B), B, 0, stream>>>(den1, 0.0f, N);
    fill_u32<<<cdiv(N, B), B, 0, stream>>>(emx1, ENC_NEG_INF, N);

    edge_max  <<<cdiv(Et, B), B, 0, stream>>>(ei, Eg, Et, asrc1, adst1, emx1);
    edge_denom<<<cdiv(Et, B), B, 0, stream>>>(ei, Eg, Et, asrc1, adst1, emx1, den1);
    edge_aggr<128><<<cdiv(Et, wavesPerBlk), B, 0, stream>>>(ei, Eg, Et, asrc1, adst1,
                                                            emx1, den1, h1, agg1);
    bias_relu<<<cdiv(N * 128, B), B, 0, stream>>>(agg1, b1, N * 128, 128);

    // ---- Layer 2 ----
    gat_gemm_wmma<32><<<cdiv(N / 16, wavesPerBlk), B, 0, stream>>>(agg1, W2, h2, N);
    gat_alpha<<<cdiv(N, wavesPerBlk), B, 0, stream>>>(h2, N, 32, as2, ad2, asrc2, adst2);

    fill_f32<<<cdiv(N * 32, B), B, 0, stream>>>(agg2, 0.0f, N * 32);
    fill_f32<<<cdiv(N, B), B, 0, stream>>>(den2, 0.0f, N);
    fill_u32<<<cdiv(N, B), B, 0, stream>>>(emx2, ENC_NEG_INF, N);

    edge_max  <<<cdiv(Et, B), B, 0, stream>>>(ei, Eg, Et, asrc2, adst2, emx2);
    edge_denom<<<cdiv(Et, B), B, 0, stream>>>(ei, Eg, Et, asrc2, adst2, emx2, den2);
    edge_aggr<32><<<cdiv(Et, wavesPerBlk), B, 0, stream>>>(ei, Eg, Et, asrc2, adst2,
                                                           emx2, den2, h2, agg2);

    logsoftmax32<<<cdiv(N, wavesPerBlk), B, 0, stream>>>(agg2, b2, out, N);
}